// KernelMultiHeadAttention_5428838662761
// MI455X (gfx1250) — compile-verified
//
#include <hip/hip_runtime.h>

// ---------------- CDNA5 WMMA f32 16x16x4 ----------------
typedef __attribute__((ext_vector_type(2))) float v2f;
typedef __attribute__((ext_vector_type(8))) float v8f;

__device__ __forceinline__ v8f wmma4(v2f a, v2f b, v8f c) {
  // (neg_a, A, neg_b, B, c_mod, C, reuse_a, reuse_b)
  return __builtin_amdgcn_wmma_f32_16x16x4_f32(false, a, false, b, (short)0, c,
                                               false, false);
}

#define NB 8
#define LL 2048
#define DM 1024
#define HD 120
#define HP 128        // padded head dim
#define NROW (NB * LL) // 16384 flat rows

// ---------------- pad q_weight [120x1024] -> [128x1024] (zeros) ----------------
__global__ void k_padw(const float* __restrict__ qw, float* __restrict__ qwp) {
  int idx = blockIdx.x * 256 + threadIdx.x;       // 131072 total
  int h = idx >> 10, d = idx & 1023;
  qwp[idx] = (h < HD) ? qw[h * DM + d] : 0.0f;
}

// ---------------- dist_weight = W @ W^T, padded to 128x128 ----------------
__global__ void k_distw(const float* __restrict__ W, float* __restrict__ dw) {
  int idx = blockIdx.x * 256 + threadIdx.x;       // 16384 total
  int h1 = idx >> 7, h2 = idx & 127;
  float acc = 0.0f;
  if (h1 < HD && h2 < HD) {
    const float* r1 = W + h1 * HD;
    const float* r2 = W + h2 * HD;
    for (int k = 0; k < HD; ++k) acc += r1[k] * r2[k];
  }
  dw[idx] = acc;
}

// ---------------- ctx = query @ qwp^T + bias  (M=16384, N=128, K=1024) ----------------
__global__ void __launch_bounds__(256) k_ctx(const float* __restrict__ q,
                                             const float* __restrict__ qwp,
                                             const float* __restrict__ qb,
                                             float* __restrict__ ctx) {
  const int lane = threadIdx.x & 31;
  const int wave = threadIdx.x >> 5;
  const int g    = lane >> 4;
  const int nlo  = lane & 15;
  const size_t i0 = ((size_t)blockIdx.x * 8 + wave) * 16;

  const float* arow = q + (i0 + (size_t)nlo) * DM; // A row m = lane&15

  v8f acc[8];
#pragma unroll
  for (int nt = 0; nt < 8; ++nt)
    acc[nt] = (v8f){0.f, 0.f, 0.f, 0.f, 0.f, 0.f, 0.f, 0.f};

#pragma unroll 4
  for (int kk = 0; kk < DM / 4; ++kk) {
    const int ko = kk * 4 + 2 * g;
    v2f a = *(const v2f*)(arow + ko);
#pragma unroll
    for (int nt = 0; nt < 8; ++nt) {
      // B[k][n] = qwp[nt*16+nlo][k]
      v2f b = *(const v2f*)(qwp + (size_t)(nt * 16 + nlo) * DM + ko);
      acc[nt] = wmma4(a, b, acc[nt]);
    }
  }

#pragma unroll
  for (int nt = 0; nt < 8; ++nt) {
    const int h = nt * 16 + nlo;
    const float bias = (h < HD) ? qb[h] : 0.0f;
#pragma unroll
    for (int r = 0; r < 8; ++r) {
      const int m = r + 8 * g;
      float v = acc[nt][r];
      v = (h < HD) ? (v + bias) : 0.0f;
      ctx[(i0 + m) * HP + h] = v;
    }
  }
}

// ---------------- trans = ctx @ dw   (M=16384, N=128, K=128) ----------------
__global__ void __launch_bounds__(256) k_trans(const float* __restrict__ ctx,
                                               const float* __restrict__ dw,
                                               float* __restrict__ trans) {
  const int lane = threadIdx.x & 31;
  const int wave = threadIdx.x >> 5;
  const int g    = lane >> 4;
  const int nlo  = lane & 15;
  const size_t i0 = ((size_t)blockIdx.x * 8 + wave) * 16;

  const float* arow = ctx + (i0 + (size_t)nlo) * HP;

  v8f acc[8];
#pragma unroll
  for (int nt = 0; nt < 8; ++nt)
    acc[nt] = (v8f){0.f, 0.f, 0.f, 0.f, 0.f, 0.f, 0.f, 0.f};

#pragma unroll
  for (int kk = 0; kk < HP / 4; ++kk) {
    const int ko = kk * 4 + 2 * g;
    v2f a = *(const v2f*)(arow + ko);
#pragma unroll
    for (int nt = 0; nt < 8; ++nt) {
      v2f b;
      b.x = dw[(size_t)ko * HP + nt * 16 + nlo];       // dw is symmetric
      b.y = dw[(size_t)(ko + 1) * HP + nt * 16 + nlo];
      acc[nt] = wmma4(a, b, acc[nt]);
    }
  }

#pragma unroll
  for (int nt = 0; nt < 8; ++nt)
#pragma unroll
    for (int r = 0; r < 8; ++r)
      trans[(i0 + r + 8 * g) * HP + nt * 16 + nlo] = acc[nt][r];
}

// ---------------- norm[i] = sum_h trans[i][h]*ctx[i][h] ----------------
__global__ void k_norm(const float* __restrict__ ctx, const float* __restrict__ trans,
                       float* __restrict__ nrm) {
  int i = blockIdx.x * 256 + threadIdx.x; // 16384 total
  const float* c = ctx + (size_t)i * HP;
  const float* t = trans + (size_t)i * HP;
  float s = 0.0f;
#pragma unroll 8
  for (int h = 0; h < HD; ++h) s += c[h] * t[h];
  nrm[i] = s;
}

// ---------------- scores + softmax, one block per (b, 16-row tile) ----------------
#define ST 2052  // LDS row stride (floats): avoids bank conflicts, 16B-aligned rows

__global__ void __launch_bounds__(256, 2) k_scores(const float* __restrict__ ctx,
                                                   const float* __restrict__ trans,
                                                   const float* __restrict__ nrm,
                                                   const float* __restrict__ pinv,
                                                   float* __restrict__ out) {
  __shared__ __align__(16) float sS[16 * ST];
  __shared__ float sNormI[16];
  __shared__ float sRed[256];
  __shared__ float sMax[16];
  __shared__ float sInv[16];

  const int b     = blockIdx.x >> 7;    // 128 row-tiles per batch
  const int itile = blockIdx.x & 127;
  const int i0    = itile * 16;
  const size_t rowbase = (size_t)b * LL;

  const int lane = threadIdx.x & 31;
  const int wave = threadIdx.x >> 5;
  const int g    = lane >> 4;
  const int nlo  = lane & 15;

  const float p    = pinv[0];
  const float coef = -0.5f * p * p;

  if (threadIdx.x < 16) sNormI[threadIdx.x] = nrm[rowbase + i0 + threadIdx.x];
  __syncthreads();

  // A fragments: trans rows i0..i0+15, K=128 -> 32 k-steps held in registers
  const float* arow = trans + (rowbase + i0 + nlo) * HP;
  v2f a[32];
#pragma unroll
  for (int kk = 0; kk < 32; ++kk) a[kk] = *(const v2f*)(arow + kk * 4 + 2 * g);

  // each wave covers 16 j-tiles of 16 columns
  for (int jt = wave; jt < 128; jt += 8) {
    const int j0 = jt * 16;
    const float* brow = ctx + (rowbase + j0 + nlo) * HP; // B[k][n] = ctx[j0+n][k]
    v8f c = (v8f){0.f, 0.f, 0.f, 0.f, 0.f, 0.f, 0.f, 0.f};
#pragma unroll
    for (int kk = 0; kk < 32; ++kk) {
      v2f bf = *(const v2f*)(brow + kk * 4 + 2 * g);
      c = wmma4(a[kk], bf, c);
    }
    const float nj = nrm[rowbase + j0 + nlo];
#pragma unroll
    for (int r = 0; r < 8; ++r) {
      const int m = r + 8 * g;
      const float dist = -2.0f * c[r] + sNormI[m] + nj;
      sS[m * ST + j0 + nlo] = __expf(coef * dist); // RBF kernel score
    }
  }
  __syncthreads();

  // softmax over j (2048) per row, 16 threads per row
  const int row  = threadIdx.x >> 4;
  const int slot = threadIdx.x & 15;

  float mx = -3.402823466e38f;
  for (int j = slot; j < LL; j += 16) mx = fmaxf(mx, sS[row * ST + j]);
  sRed[threadIdx.x] = mx;
  __syncthreads();
  if (threadIdx.x < 16) {
    float m2 = sRed[threadIdx.x * 16];
    for (int t = 1; t < 16; ++t) m2 = fmaxf(m2, sRed[threadIdx.x * 16 + t]);
    sMax[threadIdx.x] = m2;
  }
  __syncthreads();

  const float rm = sMax[row];
  float sum = 0.0f;
  for (int j = slot; j < LL; j += 16) {
    const float e = __expf(sS[row * ST + j] - rm);
    sS[row * ST + j] = e;
    sum += e;
  }
  sRed[threadIdx.x] = sum;
  __syncthreads();
  if (threadIdx.x < 16) {
    float s2 = 0.0f;
    for (int t = 0; t < 16; ++t) s2 += sRed[threadIdx.x * 16 + t];
    sInv[threadIdx.x] = 1.0f / s2;
  }
  __syncthreads();

  const float inv = sInv[row];
  float* orow = out + (rowbase + i0 + row) * LL;
  for (int j = slot * 4; j < LL; j += 64) {
    float4 v = *(const float4*)&sS[row * ST + j];
    v.x *= inv; v.y *= inv; v.z *= inv; v.w *= inv;
    *(float4*)(orow + j) = v;
  }
}

// ---------------- launch ----------------
extern "C" void kernel_launch(void* const* d_in, const int* in_sizes, int n_in,
                              void* d_out, int out_size, void* d_ws, size_t ws_size,
                              hipStream_t stream) {
  const float* q  = (const float*)d_in[0]; // [8,2048,1024]
  // d_in[1] = key (unused by reference)
  const float* qw = (const float*)d_in[2]; // [120,1024]
  const float* qb = (const float*)d_in[3]; // [120]
  const float* W  = (const float*)d_in[4]; // [120,120]
  const float* pv = (const float*)d_in[5]; // [1,1]
  float* out = (float*)d_out;              // [8,2048,2048]

  float* ws    = (float*)d_ws;
  float* qwp   = ws;                         // 128*1024      = 131072
  float* ctx   = qwp + 128 * 1024;           // 16384*128     = 2097152
  float* trans = ctx + (size_t)NROW * HP;    // 16384*128     = 2097152
  float* dw    = trans + (size_t)NROW * HP;  // 128*128       = 16384
  float* nrm   = dw + HP * HP;               // 16384
  // total ~16.7 MB of workspace

  k_padw  <<<512,  256, 0, stream>>>(qw, qwp);
  k_distw <<<64,   256, 0, stream>>>(W, dw);
  k_ctx   <<<NROW / 128, 256, 0, stream>>>(q, qwp, qb, ctx);
  k_trans <<<NROW / 128, 256, 0, stream>>>(ctx, dw, trans);
  k_norm  <<<NROW / 256, 256, 0, stream>>>(ctx, trans, nrm);
  k_scores<<<NB * (LL / 16), 256, 0, stream>>>(ctx, trans, nrm, pv, out);
}